// MultiHeadSelfAttention_61057255080357
// MI455X (gfx1250) — compile-verified
//
#include <hip/hip_runtime.h>

typedef __attribute__((ext_vector_type(2))) float v2f;
typedef __attribute__((ext_vector_type(8))) float v8f;

#define B_  2
#define S_  2048
#define D_  2048
#define H_  16
#define HD_ 128

static __device__ __forceinline__ v8f wmma4(v2f a, v2f b, v8f c) {
  // D = A(16x4 f32) * B(4x16 f32) + C(16x16 f32)
  return __builtin_amdgcn_wmma_f32_16x16x4_f32(false, a, false, b, (short)0, c, false, false);
}

// CDNA5 async global->LDS copy (ASYNCcnt-tracked, no VGPR staging).
// vdst = per-lane LDS byte address (wave-relative = low 32 bits of flat ptr),
// vaddr = per-lane 64-bit global address.
static __device__ __forceinline__ void async_load_b128(const void* g, void* l) {
  unsigned int       loff = (unsigned int)(unsigned long long)l;
  unsigned long long ga   = (unsigned long long)g;
  asm volatile("global_load_async_to_lds_b128 %0, %1, off"
               :: "v"(loff), "v"(ga) : "memory");
}
static __device__ __forceinline__ void async_wait0() {
  asm volatile("s_wait_asynccnt 0x0" ::: "memory");
}

// ---------------------------------------------------------------------------
// GEMM: C = A (MxK, row-major) * W^T  (W is NxK row-major). grid.z picks W/C.
// Block: 256 threads (8 waves). Tile: 128(M) x 128(N), K-step 16.
// Wave layout 4(m)x2(n): each wave owns 2 m-tiles x 4 n-tiles = 8 WMMA accums,
// so per k-step: 2 A-frags + 4 B-frags feed 8 WMMAs (0.75 DS frag loads/WMMA).
// ---------------------------------------------------------------------------
#define GTM 128
#define GTN 128
#define GTK 16

__global__ __launch_bounds__(256) void gemm_xwt_kernel(
    const float* __restrict__ A,
    const float* __restrict__ W0, const float* __restrict__ W1, const float* __restrict__ W2,
    float* __restrict__ C0, float* __restrict__ C1, float* __restrict__ C2,
    int M, int N, int K)
{
  __shared__ float as[GTM][GTK + 4];   // stride 80B: 16B-aligned, banks spread
  __shared__ float ws[GTN][GTK + 4];

  const float* W = (blockIdx.z == 0) ? W0 : (blockIdx.z == 1) ? W1 : W2;
  float*       C = (blockIdx.z == 0) ? C0 : (blockIdx.z == 1) ? C1 : C2;

  const int tid  = threadIdx.x;
  const int lane = tid & 31;
  const int wv   = tid >> 5;
  const int l16  = lane & 15;
  const int hi   = lane >> 4;            // 0: K=0,1  1: K=2,3
  const int wm   = wv & 3;               // m-tiles 2*wm, 2*wm+1
  const int wn   = wv >> 2;              // n-tiles 4*wn .. 4*wn+3
  const int m_blk = blockIdx.y * GTM;
  const int n_blk = blockIdx.x * GTN;

  const v8f vz = {0.f,0.f,0.f,0.f,0.f,0.f,0.f,0.f};
  v8f acc[8];
  #pragma unroll
  for (int t = 0; t < 8; ++t) acc[t] = vz;

  for (int k0 = 0; k0 < K; k0 += GTK) {
    __syncthreads();
    // A tile 128x16 + W tile 128x16: 512 float4 each, async direct to LDS
    #pragma unroll
    for (int t = 0; t < 2; ++t) {
      int i   = tid + t * 256;
      int row = i >> 2;
      int c4  = (i & 3) << 2;
      async_load_b128(&A[(size_t)(m_blk + row) * K + k0 + c4], &as[row][c4]);
      async_load_b128(&W[(size_t)(n_blk + row) * K + k0 + c4], &ws[row][c4]);
    }
    async_wait0();
    __syncthreads();
    #pragma unroll
    for (int kk = 0; kk < GTK; kk += 4) {
      v2f a0, a1;
      a0.x = as[(2 * wm + 0) * 16 + l16][kk + hi * 2];
      a0.y = as[(2 * wm + 0) * 16 + l16][kk + hi * 2 + 1];
      a1.x = as[(2 * wm + 1) * 16 + l16][kk + hi * 2];
      a1.y = as[(2 * wm + 1) * 16 + l16][kk + hi * 2 + 1];
      #pragma unroll
      for (int nt = 0; nt < 4; ++nt) {
        v2f bfr;                          // B[k][n] = W[n][k]
        bfr.x = ws[(4 * wn + nt) * 16 + l16][kk + hi * 2];
        bfr.y = ws[(4 * wn + nt) * 16 + l16][kk + hi * 2 + 1];
        acc[nt]     = wmma4(a0, bfr, acc[nt]);
        acc[4 + nt] = wmma4(a1, bfr, acc[4 + nt]);
      }
    }
  }
  // C/D layout: VGPR v, lanes 0-15 -> (M=v, N=lane); lanes 16-31 -> (M=v+8)
  #pragma unroll
  for (int g = 0; g < 2; ++g) {
    #pragma unroll
    for (int nt = 0; nt < 4; ++nt) {
      #pragma unroll
      for (int vi = 0; vi < 8; ++vi) {
        int row = m_blk + (2 * wm + g) * 16 + vi + hi * 8;
        int col = n_blk + (4 * wn + nt) * 16 + l16;
        C[(size_t)row * N + col] = acc[g * 4 + nt][vi];
      }
    }
  }
}

// ---------------------------------------------------------------------------
// RoPE applied in place to q and k. One thread per (token, head, pair-index).
// ---------------------------------------------------------------------------
__global__ void rope_kernel(float* __restrict__ q, float* __restrict__ k,
                            const int* __restrict__ pos)
{
  size_t idx = (size_t)blockIdx.x * blockDim.x + threadIdx.x;
  const size_t total = (size_t)B_ * S_ * H_ * (HD_ / 2);
  if (idx >= total) return;
  int    i   = (int)(idx & 63);
  int    h   = (int)((idx >> 6) & (H_ - 1));
  size_t tok = idx >> 10;                    // H*(HD/2) = 1024 pairs per token
  int    s   = (int)(tok & (S_ - 1));
  float  p   = (float)pos[s];
  // inv_freq = 10000^(-i/64) = 2^(-i*log2(10000)/64)
  float inv = exp2f(-(float)i * 0.20762050593046014f);
  float ang = p * inv;
  float sn, cs;
  sincosf(ang, &sn, &cs);
  size_t base = tok * (size_t)D_ + (size_t)h * HD_ + i;
  float a1 = q[base], a2 = q[base + 64];
  q[base]      = a1 * cs - a2 * sn;
  q[base + 64] = a1 * sn + a2 * cs;
  float b1 = k[base], b2 = k[base + 64];
  k[base]      = b1 * cs - b2 * sn;
  k[base + 64] = b1 * sn + b2 * cs;
}

// ---------------------------------------------------------------------------
// Flash attention: one block = 32 query rows of one (b,h); 4 waves (128 thr).
// Loop over 32-key blocks (causal). Scores and probs live only in LDS.
// QK^T and P@V use f32 WMMA. Online softmax state held by wave 0.
// ---------------------------------------------------------------------------
#define FBM 32
#define FBN 32

__global__ __launch_bounds__(128) void flash_kernel(
    const float* __restrict__ q, const float* __restrict__ k,
    const float* __restrict__ v, float* __restrict__ o)
{
  __shared__ float qs[FBM][HD_ + 4];
  __shared__ float ks[FBN][HD_ + 4];
  __shared__ float vs[FBN][HD_ + 4];
  __shared__ float ps[FBM][FBN + 2];
  __shared__ float corr[FBM];
  __shared__ float lsh[FBM];

  const int tid  = threadIdx.x;
  const int lane = tid & 31;
  const int wv   = tid >> 5;               // 0..3
  const int l16  = lane & 15;
  const int hi   = lane >> 4;
  const int qi   = blockIdx.x;
  const int h    = blockIdx.y;
  const int b    = blockIdx.z;
  const size_t bh = (size_t)b * S_ * D_ + (size_t)h * HD_;
  const int q0 = qi * FBM;

  // stage q tile (32 x 128) via async copy; first in-loop wait covers it
  for (int i = tid; i < FBM * (HD_ / 4); i += 128) {
    int row = i >> 5;
    int c4  = (i & 31) << 2;
    async_load_b128(&q[bh + (size_t)(q0 + row) * D_ + c4], &qs[row][c4]);
  }

  float m_run = -__builtin_inff();
  float l_run = 0.f;
  const int mt = wv & 1;                   // 16-row tile within 32 q rows
  const int nh = wv >> 1;                  // scores: 16-col tile; PV: 64-col half
  const v8f vz = {0.f,0.f,0.f,0.f,0.f,0.f,0.f,0.f};
  v8f oacc[4];
  oacc[0] = vz; oacc[1] = vz; oacc[2] = vz; oacc[3] = vz;
  const float scale = 0.088388347648318447f;   // 1/sqrt(128)

  for (int j = 0; j <= qi; ++j) {
    const int k0 = j * FBN;
    __syncthreads();
    for (int i = tid; i < FBN * (HD_ / 4); i += 128) {
      int row = i >> 5;
      int c4  = (i & 31) << 2;
      async_load_b128(&k[bh + (size_t)(k0 + row) * D_ + c4], &ks[row][c4]);
      async_load_b128(&v[bh + (size_t)(k0 + row) * D_ + c4], &vs[row][c4]);
    }
    async_wait0();
    __syncthreads();

    // S = q * k^T : 32x32 = 4 tiles of 16x16, one per wave
    {
      v8f sacc = vz;
      #pragma unroll
      for (int kk = 0; kk < HD_; kk += 4) {
        v2f afr;
        afr.x = qs[mt * 16 + l16][kk + hi * 2];
        afr.y = qs[mt * 16 + l16][kk + hi * 2 + 1];
        v2f bfr;                            // B[d][n] = K[n][d]
        bfr.x = ks[nh * 16 + l16][kk + hi * 2];
        bfr.y = ks[nh * 16 + l16][kk + hi * 2 + 1];
        sacc = wmma4(afr, bfr, sacc);
      }
      #pragma unroll
      for (int vi = 0; vi < 8; ++vi) {
        int r = mt * 16 + vi + hi * 8;
        int c = nh * 16 + l16;
        float val = sacc[vi] * scale;
        if (k0 + c > q0 + r) val = -__builtin_inff();   // causal mask
        ps[r][c] = val;
      }
    }
    __syncthreads();

    // online softmax: wave 0, one row per lane/thread
    if (tid < FBM) {
      float mmax = m_run;
      #pragma unroll 4
      for (int c = 0; c < FBN; ++c) mmax = fmaxf(mmax, ps[tid][c]);
      float cf = __expf(m_run - mmax);      // exp(-inf)=0 on first block
      float ls = 0.f;
      #pragma unroll 4
      for (int c = 0; c < FBN; ++c) {
        float pe = __expf(ps[tid][c] - mmax);
        ps[tid][c] = pe;
        ls += pe;
      }
      l_run = l_run * cf + ls;
      m_run = mmax;
      corr[tid] = cf;
    }
    __syncthreads();

    // rescale running O, then O += P(32x32) @ V(32x128)
    float cf8[8];
    #pragma unroll
    for (int vi = 0; vi < 8; ++vi) cf8[vi] = corr[mt * 16 + vi + hi * 8];
    #pragma unroll
    for (int t = 0; t < 4; ++t)
      #pragma unroll
      for (int vi = 0; vi < 8; ++vi) oacc[t][vi] *= cf8[vi];

    #pragma unroll
    for (int kk = 0; kk < FBN; kk += 4) {
      v2f afr;
      afr.x = ps[mt * 16 + l16][kk + hi * 2];
      afr.y = ps[mt * 16 + l16][kk + hi * 2 + 1];
      #pragma unroll
      for (int t = 0; t < 4; ++t) {
        int c0 = nh * 64 + t * 16;
        v2f bfr;                            // B[key][n] = V[key][n]
        bfr.x = vs[kk + hi * 2][c0 + l16];
        bfr.y = vs[kk + hi * 2 + 1][c0 + l16];
        oacc[t] = wmma4(afr, bfr, oacc[t]);
      }
    }
  }

  __syncthreads();
  if (tid < FBM) lsh[tid] = l_run;
  __syncthreads();
  float inv8[8];
  #pragma unroll
  for (int vi = 0; vi < 8; ++vi) inv8[vi] = 1.0f / lsh[mt * 16 + vi + hi * 8];
  #pragma unroll
  for (int t = 0; t < 4; ++t) {
    #pragma unroll
    for (int vi = 0; vi < 8; ++vi) {
      int row = q0 + mt * 16 + vi + hi * 8;
      int col = nh * 64 + t * 16 + l16;
      o[bh + (size_t)row * D_ + col] = oacc[t][vi] * inv8[vi];
    }
  }
}

// ---------------------------------------------------------------------------
extern "C" void kernel_launch(void* const* d_in, const int* in_sizes, int n_in,
                              void* d_out, int out_size, void* d_ws, size_t ws_size,
                              hipStream_t stream) {
  (void)in_sizes; (void)n_in; (void)out_size; (void)ws_size;
  const float* x  = (const float*)d_in[0];
  const float* Wq = (const float*)d_in[1];
  const float* Wk = (const float*)d_in[2];
  const float* Wv = (const float*)d_in[3];
  const float* Wo = (const float*)d_in[4];
  const int*  pos = (const int*)d_in[5];
  float* out = (float*)d_out;

  const size_t nBSD = (size_t)B_ * S_ * D_;
  float* qb = (float*)d_ws;            // also reused as the attention output O
  float* kb = qb + nBSD;
  float* vb = kb + nBSD;

  // Q/K/V projections: one launch, grid.z selects the weight matrix
  dim3 gqkv(D_ / GTN, (B_ * S_) / GTM, 3);
  gemm_xwt_kernel<<<gqkv, 256, 0, stream>>>(x, Wq, Wk, Wv, qb, kb, vb,
                                            B_ * S_, D_, D_);

  // RoPE on q and k in place
  size_t rt = (size_t)B_ * S_ * H_ * (HD_ / 2);
  rope_kernel<<<(unsigned)((rt + 255) / 256), 256, 0, stream>>>(qb, kb, pos);

  // Flash attention: O written over the q buffer (each block reads exactly
  // the q region it later writes, so the alias is safe)
  dim3 gf(S_ / FBM, H_, B_);
  flash_kernel<<<gf, 128, 0, stream>>>(qb, kb, vb, qb);

  // Output projection -> d_out
  dim3 go(D_ / GTN, (B_ * S_) / GTM, 1);
  gemm_xwt_kernel<<<go, 256, 0, stream>>>(qb, Wo, Wo, Wo, out, out, out,
                                          B_ * S_, D_, D_);
}